// CriticalityLoss_4595615006787
// MI455X (gfx1250) — compile-verified
//
#include <hip/hip_runtime.h>
#include <math.h>

// ---------------------------------------------------------------------------
// CriticalityLoss for MI455X (gfx1250, wave32)
//
// total = MSE(pred[:,0],target[:,0])
//       + 0.5 * MSE(pred[:,1:],target[:,1:])
//       + 0.3 * ( mean(clse) - mean(s) )
// where s = pred[:,0] permuted by stable argsort(-target[:,0]) and
// clse = reverse cumulative logsumexp of s.
//
// Memory-bound problem (~0.75 GB total traffic ~ 32us @ 23.3 TB/s), dominated
// by a stable 4M-element key/value radix sort. WMMA f32 16x16x4 is used as a
// co-executing reducer for the 7-column squared-diff sum (B = ones fragment);
// global_prefetch_b8 feeds the streaming phases.
// ---------------------------------------------------------------------------

#define MSE_GRID     1024
#define MSE_BLK      256
#define SORT_BLOCKS  256   // radix: 256 blocks x 256 threads, 8-bit digits
#define SCAN_BLOCKS  1024  // suffix-scan phase
#define SCAN_THREADS 256

typedef __attribute__((ext_vector_type(2))) float v2f;
typedef __attribute__((ext_vector_type(8))) float v8f;

#if defined(__gfx1250__) && __has_builtin(__builtin_amdgcn_wmma_f32_16x16x4_f32)
#define USE_WMMA_REDUCE 1
#else
#define USE_WMMA_REDUCE 0
#endif

// ---- float <-> order-preserving uint encodings -----------------------------
// ascending key: larger float -> larger key (for atomicMax of s)
__device__ __forceinline__ unsigned enc_asc(float f) {
    unsigned u = __float_as_uint(f);
    return (u & 0x80000000u) ? ~u : (u | 0x80000000u);
}
__device__ __forceinline__ float dec_asc(unsigned k) {
    unsigned u = (k & 0x80000000u) ? (k ^ 0x80000000u) : ~k;
    return __uint_as_float(u);
}
// descending key: ascending uint sort of key == descending float sort
__device__ __forceinline__ unsigned enc_desc(float f) {
    unsigned u = __float_as_uint(f);
    return (u & 0x80000000u) ? u : (~u & 0x7FFFFFFFu);
}

// ---- per-row evaluation ----------------------------------------------------
__device__ __forceinline__ void row_eval(const float* __restrict__ pred,
                                         const float* __restrict__ targ,
                                         int i, float& d0sq, float& r,
                                         float& s, unsigned& key) {
    const float4* pr = (const float4*)(pred + (size_t)i * 8);
    const float4* tr = (const float4*)(targ + (size_t)i * 8);
    float4 p0 = pr[0], p1 = pr[1];
    float4 t0 = tr[0], t1 = tr[1];
    float d0 = p0.x - t0.x;
    d0sq = d0 * d0;
    float a1 = p0.y - t0.y, a2 = p0.z - t0.z, a3 = p0.w - t0.w;
    float a4 = p1.x - t1.x, a5 = p1.y - t1.y, a6 = p1.z - t1.z, a7 = p1.w - t1.w;
    r = a1*a1 + a2*a2 + a3*a3 + a4*a4 + a5*a5 + a6*a6 + a7*a7;
    s = p0.x;
    key = enc_desc(t0.x);
}

// ---------------------------------------------------------------------------
// init: zero accumulators (harness poisons ws; graph replays need reset)
// sums[0]=sum d0^2, sums[1]=sum rmav, sums[2]=sum s, sums[3]=sum clse
// ---------------------------------------------------------------------------
__global__ void k_init(double* sums, unsigned* maxKey) {
    if (threadIdx.x < 4) sums[threadIdx.x] = 0.0;
    if (threadIdx.x == 0) *maxKey = 0u;
}

// ---------------------------------------------------------------------------
// Fused: MSE partial sums + max(s) + emit (key,val) pairs for the sort.
// The rmav squared-diff stream is reduced through V_WMMA_F32_16X16X4_F32:
// A = packed diffs^2 (2 rows/lane per issue), B = ones => C accumulates row
// sums; each of the 16 C columns carries the full sum, so
// sum_lanes(sum_v c[v]) == 16 * true_sum.
// ---------------------------------------------------------------------------
__global__ void k_fuse_mse_keys(const float* __restrict__ pred,
                                const float* __restrict__ targ,
                                unsigned* __restrict__ keys,
                                float* __restrict__ vals,
                                double* __restrict__ sums,
                                unsigned* __restrict__ maxKey,
                                int n) {
    const int tid = threadIdx.x;
    const int gid = blockIdx.x * blockDim.x + tid;
    const int stride = gridDim.x * blockDim.x;

    double z = 0.0, ssum = 0.0;
    float smaxLane = -3.402823466e38f;
    float rTail = 0.0f;

#if USE_WMMA_REDUCE
    v8f c = {0.f, 0.f, 0.f, 0.f, 0.f, 0.f, 0.f, 0.f};
    v2f bOnes; bOnes.x = 1.0f; bOnes.y = 1.0f;
    float pendR = 0.0f;
    int   pend  = 0;          // uniform across the wave (same trip count)
#else
    double rAcc = 0.0;
#endif

    const int full = n / stride;  // uniform => EXEC all-ones inside this loop
    for (int k = 0; k < full; ++k) {
        const int i = gid + k * stride;
        // speculative prefetch ~4 iterations ahead (OOB silently dropped)
        __builtin_prefetch(pred + ((size_t)i + (size_t)stride * 4) * 8, 0, 0);
        __builtin_prefetch(targ + ((size_t)i + (size_t)stride * 4) * 8, 0, 0);

        float d0sq, r, s; unsigned key;
        row_eval(pred, targ, i, d0sq, r, s, key);
        z += (double)d0sq;
        ssum += (double)s;
        smaxLane = fmaxf(smaxLane, s);
        keys[i] = key;
        vals[i] = s;
#if USE_WMMA_REDUCE
        if (!pend) { pendR = r; pend = 1; }
        else {
            v2f a; a.x = pendR; a.y = r;
            c = __builtin_amdgcn_wmma_f32_16x16x4_f32(
                    false, a, false, bOnes, (short)0, c, false, false);
            pend = 0;
        }
#else
        rAcc += (double)r;
#endif
    }
    // tail (possibly divergent -> handled scalar, outside WMMA)
    {
        const int i = gid + full * stride;
        if (i < n) {
            float d0sq, r, s; unsigned key;
            row_eval(pred, targ, i, d0sq, r, s, key);
            z += (double)d0sq;
            ssum += (double)s;
            smaxLane = fmaxf(smaxLane, s);
            keys[i] = key;
            vals[i] = s;
            rTail += r;
        }
    }
#if USE_WMMA_REDUCE
    if (pend) {  // uniform flush
        v2f a; a.x = pendR; a.y = 0.0f;
        c = __builtin_amdgcn_wmma_f32_16x16x4_f32(
                false, a, false, bOnes, (short)0, c, false, false);
    }
    double rLane = (double)(c[0] + c[1] + c[2] + c[3] +
                            c[4] + c[5] + c[6] + c[7]) * (1.0 / 16.0)
                   + (double)rTail;
#else
    double rLane = rAcc + (double)rTail;
#endif

    __shared__ double sZ[MSE_BLK], sR[MSE_BLK], sS[MSE_BLK];
    __shared__ unsigned sM[MSE_BLK];
    sZ[tid] = z; sR[tid] = rLane; sS[tid] = ssum; sM[tid] = enc_asc(smaxLane);
    __syncthreads();
    for (int o = MSE_BLK / 2; o > 0; o >>= 1) {
        if (tid < o) {
            sZ[tid] += sZ[tid + o];
            sR[tid] += sR[tid + o];
            sS[tid] += sS[tid + o];
            sM[tid] = sM[tid] > sM[tid + o] ? sM[tid] : sM[tid + o];
        }
        __syncthreads();
    }
    if (tid == 0) {
        atomicAdd(&sums[0], sZ[0]);
        atomicAdd(&sums[1], sR[0]);
        atomicAdd(&sums[2], sS[0]);
        atomicMax(maxKey, sM[0]);
    }
}

// ---------------------------------------------------------------------------
// Radix sort (LSD, 4 x 8-bit, stable) on (key,val): hist -> scan -> scatter.
// hist layout digit-major: hist[d*SORT_BLOCKS + b].
// ---------------------------------------------------------------------------
__global__ void k_radix_hist(const unsigned* __restrict__ keys,
                             unsigned* __restrict__ hist, int n, int shift) {
    __shared__ unsigned cnt[256];
    const int t = threadIdx.x, b = blockIdx.x;
    cnt[t] = 0;
    __syncthreads();
    const int tile = (n + gridDim.x - 1) / gridDim.x;
    const int beg = b * tile;
    int end = beg + tile; if (end > n) end = n;
    for (int i = beg + t; i < end; i += blockDim.x) {
        __builtin_prefetch(keys + i + blockDim.x * 8, 0, 0);
        atomicAdd(&cnt[(keys[i] >> shift) & 0xFFu], 1u);
    }
    __syncthreads();
    hist[(unsigned)t * gridDim.x + b] = cnt[t];
}

// single block, 1024 threads: exclusive scan over 256*SORT_BLOCKS entries
__global__ void k_radix_scan(unsigned* __restrict__ hist) {
    const int t = threadIdx.x;               // 1024 threads
    const int SEG = (256 * SORT_BLOCKS) / 1024;  // 64
    const int base = t * SEG;
    unsigned run = 0;
    for (int j = 0; j < SEG; ++j) {          // local exclusive scan
        unsigned v = hist[base + j];
        hist[base + j] = run;
        run += v;
    }
    __shared__ unsigned st[1024];
    st[t] = run;
    __syncthreads();
    for (int off = 1; off < 1024; off <<= 1) {   // inclusive Hillis-Steele
        unsigned v = (t >= off) ? st[t - off] : 0u;
        __syncthreads();
        st[t] += v;
        __syncthreads();
    }
    const unsigned offv = st[t] - run;       // exclusive segment offset
    for (int j = 0; j < SEG; ++j) hist[base + j] += offv;
}

// stable scatter: in-block ranking via wave32 ballots + LDS per-wave counts
__global__ void k_radix_scatter(const unsigned* __restrict__ keysIn,
                                const float* __restrict__ valsIn,
                                unsigned* __restrict__ keysOut,
                                float* __restrict__ valsOut,
                                const unsigned* __restrict__ hist,
                                int n, int shift) {
    __shared__ unsigned base[256];
    __shared__ unsigned wc[8][256];
    const int t = threadIdx.x, b = blockIdx.x;
    const int lane = t & 31, wave = t >> 5;
    base[t] = hist[(unsigned)t * gridDim.x + b];
    const int tile = (n + gridDim.x - 1) / gridDim.x;
    const int beg = b * tile;
    int end = beg + tile; if (end > n) end = n;

    for (int cs = beg; cs < end; cs += blockDim.x) {
        for (int w = 0; w < 8; ++w) wc[w][t] = 0;
        __syncthreads();

        const int i = cs + t;
        const bool valid = i < end;
        unsigned key = 0, dig = 0; float val = 0.0f;
        if (valid) { key = keysIn[i]; val = valsIn[i]; dig = (key >> shift) & 0xFFu; }

        const unsigned vm = (unsigned)__ballot(valid ? 1 : 0);
        unsigned mask = vm;
        for (int bit = 0; bit < 8; ++bit) {
            unsigned bal = (unsigned)__ballot((int)((dig >> bit) & 1u));
            mask &= ((dig >> bit) & 1u) ? bal : ~bal;
        }
        const unsigned rank = __popc(mask & ((1u << lane) - 1u));
        if (valid && rank == 0) wc[wave][dig] = __popc(mask);
        __syncthreads();

        if (valid) {
            unsigned pre = 0;
            for (int w = 0; w < wave; ++w) pre += wc[w][dig];
            const unsigned pos = base[dig] + pre + rank;
            keysOut[pos] = key;
            valsOut[pos] = val;
        }
        __syncthreads();
        unsigned tot = 0;
        for (int w = 0; w < 8; ++w) tot += wc[w][t];
        base[t] += tot;
        __syncthreads();
    }
}

// ---------------------------------------------------------------------------
// Reverse cumulative logsumexp: 3 phases, fp64 accumulation.
// ---------------------------------------------------------------------------
__global__ void k_expsum(const float* __restrict__ s,
                         const unsigned* __restrict__ maxKey,
                         double* __restrict__ blkSum, int n) {
    const double m = (double)dec_asc(*maxKey);
    const int tile = (n + gridDim.x - 1) / gridDim.x;
    const int beg = blockIdx.x * tile;
    int end = beg + tile; if (end > n) end = n;
    double acc = 0.0;
    for (int i = beg + threadIdx.x; i < end; i += blockDim.x) {
        __builtin_prefetch(s + i + blockDim.x * 8, 0, 0);
        acc += exp((double)s[i] - m);
    }
    __shared__ double sh[SCAN_THREADS];
    sh[threadIdx.x] = acc;
    __syncthreads();
    for (int o = SCAN_THREADS / 2; o > 0; o >>= 1) {
        if (threadIdx.x < o) sh[threadIdx.x] += sh[threadIdx.x + o];
        __syncthreads();
    }
    if (threadIdx.x == 0) blkSum[blockIdx.x] = sh[0];
}

// single block of SCAN_BLOCKS threads: blkSuffix[b] = sum_{b'>b} blkSum[b']
__global__ void k_suffix(const double* __restrict__ blkSum,
                         double* __restrict__ blkSuffix, int nb) {
    __shared__ double sc[SCAN_BLOCKS];
    const int t = threadIdx.x;
    sc[t] = (t < nb) ? blkSum[nb - 1 - t] : 0.0;
    __syncthreads();
    const double own = sc[t];
    for (int off = 1; off < SCAN_BLOCKS; off <<= 1) {
        double v = (t >= off) ? sc[t - off] : 0.0;
        __syncthreads();
        sc[t] += v;
        __syncthreads();
    }
    if (t < nb) blkSuffix[nb - 1 - t] = sc[t] - own;
}

// per-block reverse inclusive scan over exp(s-m), sum of clse = m + log(S_i)
__global__ void k_clse(const float* __restrict__ s,
                       const unsigned* __restrict__ maxKey,
                       const double* __restrict__ blkSuffix,
                       double* __restrict__ sums, int n) {
    const double m = (double)dec_asc(*maxKey);
    const int tile = (n + gridDim.x - 1) / gridDim.x;
    const int beg = blockIdx.x * tile;
    int end = beg + tile; if (end > n) end = n;
    const int nchunks = (end > beg) ? (end - beg + SCAN_THREADS - 1) / SCAN_THREADS : 0;
    double carry = blkSuffix[blockIdx.x];
    double acc = 0.0;
    __shared__ double sc[SCAN_THREADS];
    const int t = threadIdx.x;

    for (int c = nchunks - 1; c >= 0; --c) {
        const int cs = beg + c * SCAN_THREADS;
        const int idx = cs + (SCAN_THREADS - 1 - t);  // reversed mapping
        const bool valid = idx < end;
        const double e = valid ? exp((double)s[idx] - m) : 0.0;
        sc[t] = e;
        __syncthreads();
        for (int off = 1; off < SCAN_THREADS; off <<= 1) {
            double v = (t >= off) ? sc[t - off] : 0.0;
            __syncthreads();
            sc[t] += v;
            __syncthreads();
        }
        if (valid) acc += m + log(sc[t] + carry);   // suffix sum incl. self
        const double tot = sc[SCAN_THREADS - 1];
        __syncthreads();
        carry += tot;
    }

    sc[t] = acc;
    __syncthreads();
    for (int o = SCAN_THREADS / 2; o > 0; o >>= 1) {
        if (t < o) sc[t] += sc[t + o];
        __syncthreads();
    }
    if (t == 0) atomicAdd(&sums[3], sc[0]);
}

// ---------------------------------------------------------------------------
__global__ void k_finalize(const double* __restrict__ sums,
                           float* __restrict__ out, int n) {
    const double N = (double)n;
    const double mse0 = sums[0] / N;
    const double mrm  = sums[1] / (N * 7.0);
    const double rank = (sums[3] - sums[2]) / N;   // mean(clse) - mean(s)
    out[0] = (float)(mse0 + 0.5 * mrm + 0.3 * rank);
}

// ---------------------------------------------------------------------------
extern "C" void kernel_launch(void* const* d_in, const int* in_sizes, int n_in,
                              void* d_out, int out_size, void* d_ws, size_t ws_size,
                              hipStream_t stream) {
    const float* pred = (const float*)d_in[0];
    const float* targ = (const float*)d_in[1];
    float* out = (float*)d_out;
    const int D = 8;
    const int n = in_sizes[0] / D;   // rows

    // workspace layout (needs ~16*n + 0.3 MB bytes)
    char* ws = (char*)d_ws;
    double*   sums   = (double*)ws;                 // 4 doubles
    unsigned* maxKey = (unsigned*)(ws + 32);
    size_t off = 64;
    unsigned* keysA = (unsigned*)(ws + off); off += (size_t)n * 4;
    float*    valsA = (float*)(ws + off);    off += (size_t)n * 4;
    unsigned* keysB = (unsigned*)(ws + off); off += (size_t)n * 4;
    float*    valsB = (float*)(ws + off);    off += (size_t)n * 4;
    unsigned* hist  = (unsigned*)(ws + off); off += (size_t)256 * SORT_BLOCKS * 4;
    double*   blkSum = (double*)(ws + off);  off += (size_t)SCAN_BLOCKS * 8;
    double*   blkSuf = (double*)(ws + off);  off += (size_t)SCAN_BLOCKS * 8;
    (void)ws_size; (void)n_in; (void)out_size;

    k_init<<<1, 64, 0, stream>>>(sums, maxKey);
    k_fuse_mse_keys<<<MSE_GRID, MSE_BLK, 0, stream>>>(pred, targ, keysA, valsA,
                                                      sums, maxKey, n);

    unsigned* ki = keysA; float* vi = valsA;
    unsigned* ko = keysB; float* vo = valsB;
    for (int shift = 0; shift < 32; shift += 8) {
        k_radix_hist<<<SORT_BLOCKS, 256, 0, stream>>>(ki, hist, n, shift);
        k_radix_scan<<<1, 1024, 0, stream>>>(hist);
        k_radix_scatter<<<SORT_BLOCKS, 256, 0, stream>>>(ki, vi, ko, vo, hist, n, shift);
        unsigned* tk = ki; ki = ko; ko = tk;
        float*    tv = vi; vi = vo; vo = tv;
    }
    // after 4 passes, sorted (key ascending == target descending) is in ki/vi

    k_expsum<<<SCAN_BLOCKS, SCAN_THREADS, 0, stream>>>(vi, maxKey, blkSum, n);
    k_suffix<<<1, SCAN_BLOCKS, 0, stream>>>(blkSum, blkSuf, SCAN_BLOCKS);
    k_clse<<<SCAN_BLOCKS, SCAN_THREADS, 0, stream>>>(vi, maxKey, blkSuf, sums, n);
    k_finalize<<<1, 1, 0, stream>>>(sums, out, n);
}